// CRF_32495722562252
// MI455X (gfx1250) — compile-verified
//
#include <hip/hip_runtime.h>
#include <hip/hip_bf16.h>

typedef __attribute__((ext_vector_type(2))) float v2f;
typedef __attribute__((ext_vector_type(8))) float v8f;

#define T_LEN 512
#define B_SZ  256
#define N_ST  128

// workspace layout (float offsets)
#define WS_LOGZ  0       // 16 floats  : per-forward-block logZ partials
#define WS_SCORE 16      // 256 floats : per-batch score partials
#define WS_MT    272     // 128 floats : mT[j] = max_i trans[i][j]
#define WS_ETR   400     // 128*128    : E[i][j] = exp(trans[i][j] - mT[j])

// ---------------------------------------------------------------------------
// Kernel 1: precompute column maxes of trans and E = exp(trans - mT)
// ---------------------------------------------------------------------------
__global__ void crf_prep(const float* __restrict__ trans, float* __restrict__ ws) {
    int j = threadIdx.x;                       // 128 threads, 1 block
    float m = trans[j];
    for (int i = 1; i < N_ST; ++i) m = fmaxf(m, trans[i * N_ST + j]);
    ws[WS_MT + j] = m;
    float* E = ws + WS_ETR;
    for (int i = 0; i < N_ST; ++i)
        E[i * N_ST + j] = __expf(trans[i * N_ST + j] - m);
}

// ---------------------------------------------------------------------------
// Kernel 2: forward recursion. 16 blocks x 256 threads (8 wave32).
// Block handles 16 batch rows; wave w handles state columns [16w, 16w+16).
// Per step: P = exp(alpha - mA)  (LDS), then S = P x E via chained
// V_WMMA_F32_16X16X4_F32 (E slab pinned in VGPRs), then
// alpha[b][j] = log(S) + mA[b] + mT[j] + emit[t][b][j] where mask[t][b].
// ---------------------------------------------------------------------------
__launch_bounds__(256)
__global__ void crf_forward(const float* __restrict__ emit,
                            const unsigned char* __restrict__ mask,
                            const float* __restrict__ strans,
                            const float* __restrict__ etrans,
                            float* __restrict__ ws) {
    __shared__ float alpha[16][130];   // padded rows: stride 130 floats
    __shared__ float P[16][130];
    __shared__ float red[16][16];
    __shared__ float mA[16];
    __shared__ float mk[16];

    const int tid  = threadIdx.x;
    const int wg   = blockIdx.x;          // 0..15
    const int b0   = wg * 16;             // first batch row of this block
    const int lane = tid & 31;
    const int wv   = tid >> 5;            // wave id 0..7
    const int j0   = wv * 16;             // state-column tile base
    const int r    = lane & 15;           // A row / B-C column within tile
    const int koff = (lane >> 4) * 2;     // K sub-offset per 16-lane half
    const int jc   = j0 + r;              // global state column of this lane

    // --- pin this wave's 128x16 slab of E in registers (64 VGPRs, reused all T)
    const float* E = ws + WS_ETR;
    v2f bfrag[32];
#pragma unroll
    for (int kk = 0; kk < 32; ++kk) {
        int k = kk * 4 + koff;
        bfrag[kk].x = E[(k    ) * N_ST + jc];
        bfrag[kk].y = E[(k + 1) * N_ST + jc];
    }
    const float mTj = ws[WS_MT + jc];

    // --- alpha0 = strans + emit[0]
    {
        int b = tid >> 4, i0 = (tid & 15) * 8;
#pragma unroll
        for (int q = 0; q < 8; ++q) {
            int i = i0 + q;
            alpha[b][i] = strans[i] + emit[(size_t)(b0 + b) * N_ST + i];
        }
    }
    __syncthreads();

    for (int t = 1; t < T_LEN; ++t) {
        // phase 1: row maxes mA[b], mask flags, P = exp(alpha - mA)
        int b = tid >> 4, seg = tid & 15, i0 = seg * 8;
        float m = alpha[b][i0];
#pragma unroll
        for (int q = 1; q < 8; ++q) m = fmaxf(m, alpha[b][i0 + q]);
        red[b][seg] = m;
        __syncthreads();
        if (tid < 16) {
            float mm = red[tid][0];
#pragma unroll
            for (int q = 1; q < 16; ++q) mm = fmaxf(mm, red[tid][q]);
            mA[tid] = mm;
            mk[tid] = (float)mask[t * B_SZ + b0 + tid];
        }
        __syncthreads();
        float mb = mA[b];
#pragma unroll
        for (int q = 0; q < 8; ++q) {
            int i = i0 + q;
            P[b][i] = __expf(alpha[b][i] - mb);
        }
        __syncthreads();

        // phase 2: S(16x16) = P(16x128) x E(128x16), 32 chained f32 WMMAs
        v8f c = {};
#pragma unroll
        for (int kk = 0; kk < 32; ++kk) {
            int k = kk * 4 + koff;
            v2f a = *(const v2f*)&P[r][k];   // 8B-aligned (row stride 520B, k even)
            c = __builtin_amdgcn_wmma_f32_16x16x4_f32(
                    false, a, false, bfrag[kk], (short)0, c, false, false);
        }

        // phase 3: masked alpha update (mask is per batch row)
        const float* em = emit + (size_t)t * B_SZ * N_ST + (size_t)b0 * N_ST;
#pragma unroll
        for (int v = 0; v < 8; ++v) {
            int row = v + ((lane >> 4) << 3);     // C/D layout: M = v (+8 hi half)
            if (mk[row] != 0.0f) {
                alpha[row][jc] = __logf(c[v]) + mA[row] + mTj
                               + em[row * N_ST + jc];
            }
        }
        __syncthreads();
    }

    // logZ partial for this block: sum_b logsumexp_i(alpha[b][i] + etrans[i])
    if (tid < 16) {
        int b = tid;
        float m = alpha[b][0] + etrans[0];
        for (int i = 1; i < N_ST; ++i) m = fmaxf(m, alpha[b][i] + etrans[i]);
        float s = 0.f;
        for (int i = 0; i < N_ST; ++i) s += __expf(alpha[b][i] + etrans[i] - m);
        red[0][tid] = __logf(s) + m;
    }
    __syncthreads();
    if (tid == 0) {
        float s = 0.f;
#pragma unroll
        for (int q = 0; q < 16; ++q) s += red[0][q];
        ws[WS_LOGZ + wg] = s;
    }
}

// ---------------------------------------------------------------------------
// Kernel 3: gold-path score. One block per batch element b (256 x 256).
// ---------------------------------------------------------------------------
__global__ void crf_score(const float* __restrict__ emit,
                          const int* __restrict__ target,
                          const unsigned char* __restrict__ mask,
                          const float* __restrict__ trans,
                          const float* __restrict__ strans,
                          const float* __restrict__ etrans,
                          float* __restrict__ ws) {
    __shared__ float sred[256];
    __shared__ int   cred[256];
    const int b = blockIdx.x, tid = threadIdx.x;

    float s = 0.f; int cnt = 0;
    for (int t = tid; t < T_LEN; t += 256) {
        int m = mask[t * B_SZ + b];
        cnt += m;
        if (m) {
            int tgt = target[t * B_SZ + b];
            float v = emit[((size_t)t * B_SZ + b) * N_ST + tgt];
            if (t > 0) v += trans[target[(t - 1) * B_SZ + b] * N_ST + tgt];
            s += v;
        }
    }
    sred[tid] = s; cred[tid] = cnt;
    __syncthreads();
    for (int off = 128; off > 0; off >>= 1) {
        if (tid < off) { sred[tid] += sred[tid + off]; cred[tid] += cred[tid + off]; }
        __syncthreads();
    }
    if (tid == 0) {
        int ends = cred[0] - 1;                              // lengths[b] - 1
        ws[WS_SCORE + b] = sred[0] + strans[target[b]]       // target[0][b]
                         + etrans[target[ends * B_SZ + b]];
    }
}

// ---------------------------------------------------------------------------
// Kernel 4: final deterministic reduction -> out[0] = (logZ - score)/B
// ---------------------------------------------------------------------------
__global__ void crf_final(const float* __restrict__ ws, float* __restrict__ out) {
    __shared__ float sred[256];
    int tid = threadIdx.x;
    sred[tid] = ws[WS_SCORE + tid];
    __syncthreads();
    for (int off = 128; off > 0; off >>= 1) {
        if (tid < off) sred[tid] += sred[tid + off];
        __syncthreads();
    }
    if (tid == 0) {
        float lz = 0.f;
#pragma unroll
        for (int q = 0; q < 16; ++q) lz += ws[WS_LOGZ + q];
        out[0] = (lz - sred[0]) / (float)B_SZ;
    }
}

// ---------------------------------------------------------------------------
extern "C" void kernel_launch(void* const* d_in, const int* in_sizes, int n_in,
                              void* d_out, int out_size, void* d_ws, size_t ws_size,
                              hipStream_t stream) {
    const float*         emit   = (const float*)d_in[0];          // (T,B,N) f32
    const int*           target = (const int*)d_in[1];            // (T,B) i32
    const unsigned char* mask   = (const unsigned char*)d_in[2];  // (T,B) bool (1B)
    const float*         trans  = (const float*)d_in[3];          // (N,N) f32
    const float*         strans = (const float*)d_in[4];          // (N,) f32
    const float*         etrans = (const float*)d_in[5];          // (N,) f32
    float* out = (float*)d_out;
    float* ws  = (float*)d_ws;   // needs ~67 KB

    crf_prep   <<<1,   128, 0, stream>>>(trans, ws);
    crf_forward<<<16,  256, 0, stream>>>(emit, mask, strans, etrans, ws);
    crf_score  <<<256, 256, 0, stream>>>(emit, target, mask, trans, strans, etrans, ws);
    crf_final  <<<1,   256, 0, stream>>>(ws, out);
}